// LongTermLSTM_40389872452133
// MI455X (gfx1250) — compile-verified
//
#include <hip/hip_runtime.h>

// LSTM last-step kernel for MI455X (gfx1250), fp32 WMMA 16x16x4 path, v3.
//
// One workgroup = 256 threads = 8 waves = TWO 16-sequence tiles sharing one
// LDS weight image (the weight image is the LDS hog; sharing it doubles
// waves/SIMD from 1 to 2 so the ds_load -> v_wmma latency chain can be hidden
// by the co-resident wave). Wave (pair p, gate g) computes gate g for pair
// p's 16 sequences.
//
// Per-step GEMM is a single augmented matmul:
//   state vector per sequence = [h(100), 1.0 (bias row), pad(3), x_t(4)] (K=108)
//   weight image rows          = [W_hh  , b_ih+b_hh    , 0     , W_ih  ]
// so gates = state @ W_aug^T in 27 K-iterations of v_wmma_f32_16x16x4_f32
// with zero-initialized accumulators (inline C=0). Gate columns padded
// 100->112 so each gate is exactly 7 N-tiles. State lives in LDS in the WMMA
// A-fragment layout; c lives in registers; per-sequence length masking
// freezes h and c exactly like the reference.

#define HIDDEN  100
#define INPUT   4
#define T_MAX   256
#define HP      112         // per-gate padded width (7 tiles of 16)
#define NCOLS   448         // 4 * HP
#define NTILES  28          // NCOLS / 16
#define TPW     7           // tiles per wave (one gate per wave)
#define KP      108         // padded K: 100 h + 1 bias + 3 pad + 4 x
#define KITERS  27          // KP / 4
#define SEQ_PER_WG 32       // two 16-sequence pairs

#define G_SZ    (4 * 16 * HP)        // 7168 floats per pair
#define HA_SZ   (KITERS * 64)        // 1728 floats per pair

// LDS layout (float offsets)
#define WB_OFF   0                        // W_aug B-frags [27][28][32][2] = 48384
#define G_OFF    (WB_OFF + 48384)         // gates, 2 pairs  = 14336
#define HA_OFF   (G_OFF + 2 * G_SZ)       // state, 2 pairs  = 3456
#define LEN_OFF  (HA_OFF + 2 * HA_SZ)     // lengths [32] (as int)
#define SMEM_FLOATS (LEN_OFF + 32)        // 66208 floats = 264832 bytes

typedef __attribute__((ext_vector_type(2))) float v2f;
typedef __attribute__((ext_vector_type(4))) float v4f;
typedef __attribute__((ext_vector_type(8))) float v8f;

__device__ __forceinline__ float sigmoid_fast(float x) {
    return __builtin_amdgcn_rcpf(1.0f + __expf(-x));
}
__device__ __forceinline__ float tanh_fast(float x) {
    float e = __expf(2.0f * x);
    return 1.0f - 2.0f * __builtin_amdgcn_rcpf(e + 1.0f);
}

// Address of state[s][u] inside one pair's A-fragment-layout buffer:
// K-iter k4 = u/4; lanes 0-15 hold K = 4k4+{0,1}, lanes 16-31 hold 4k4+{2,3}.
__device__ __forceinline__ int hA_addr(int s, int u) {
    int k4 = u >> 2;
    int r  = u & 3;
    int ln = s + ((r & 2) ? 16 : 0);
    return (k4 * 32 + ln) * 2 + (r & 1);
}

__global__ __launch_bounds__(256)
void LongTermLSTM_40389872452133_kernel(
    const float* __restrict__ x,       // [4096, 256, 4]
    const float* __restrict__ w_ih,    // [400, 4]
    const float* __restrict__ w_hh,    // [400, 100]
    const float* __restrict__ b_ih,    // [400]
    const float* __restrict__ b_hh,    // [400]
    const int*   __restrict__ lengths, // [4096]
    float*       __restrict__ out)     // [4096, 100]
{
    extern __shared__ float smem[];
    float* wB   = smem + WB_OFF;
    int*   slen = (int*)(smem + LEN_OFF);

    const int tid      = threadIdx.x;
    const int lane     = tid & 31;
    const int wave     = (tid >> 5) & 3;    // gate id (i,f,g,o)
    const int pair     = tid >> 7;          // which 16-sequence tile
    const int seq_base = blockIdx.x * SEQ_PER_WG;

    float* Gp  = smem + G_OFF + pair * G_SZ;    // this pair's gate buffer
    float* hAp = smem + HA_OFF + pair * HA_SZ;  // this pair's state buffer

    // ---------------- prologue: build augmented weight image in LDS --------
    // Padded column c in [0,448): gate g = c/112, unit u = c%112 (u>=100 -> 0).
    // Augmented K row k in [0,108): 0..99 = W_hh[col][k], 100 = bias[col],
    // 101..103 = 0, 104..107 = W_ih[col][k-104].
    // B-frag swizzle: lane = (c%16) + 16*(k%4 >= 2), dword j = k&1.
    for (int idx = tid; idx < NCOLS * KP; idx += 256) {
        int c = idx / KP;
        int k = idx - c * KP;
        int g = c / HP, u = c - g * HP;
        float v = 0.0f;
        if (u < HIDDEN) {
            int row = g * HIDDEN + u;
            if (k < HIDDEN)        v = w_hh[row * HIDDEN + k];
            else if (k == HIDDEN)  v = b_ih[row] + b_hh[row];
            else if (k >= 104)     v = w_ih[row * INPUT + (k - 104)];
        }
        int tt = c >> 4, lc = c & 15;
        int k4 = k >> 2, r = k & 3;
        int ln = lc + ((r & 2) ? 16 : 0);
        wB[((k4 * NTILES + tt) * 32 + ln) * 2 + (r & 1)] = v;
    }
    for (int i = tid; i < 2 * HA_SZ; i += 256)
        smem[HA_OFF + i] = 0.0f;                       // h0 = 0 (both pairs)
    if (tid < SEQ_PER_WG) slen[tid] = lengths[seq_base + tid];
    __syncthreads();

    // State slots: bias-constant row (u=100) and x_0 (u=104..107).
    if (tid < SEQ_PER_WG) {
        int p = tid >> 4, s = tid & 15;
        float* hAq = smem + HA_OFF + p * HA_SZ;
        hAq[hA_addr(s, 100)] = 1.0f;
        v4f xv = *(const v4f*)(x + (size_t)(seq_base + tid) * T_MAX * INPUT);
        hAq[(26 * 32 + s) * 2 + 0]        = xv.x;   // u=104
        hAq[(26 * 32 + s) * 2 + 1]        = xv.y;   // u=105
        hAq[(26 * 32 + (s + 16)) * 2 + 0] = xv.z;   // u=106
        hAq[(26 * 32 + (s + 16)) * 2 + 1] = xv.w;   // u=107
    }

    int maxlen = 0;
    for (int s = 0; s < SEQ_PER_WG; ++s) maxlen = max(maxlen, slen[s]);

    // Hoisted, time-invariant elementwise indexing for e = tid + 256*r,
    // e < 3200: global seq sg = e/100, pair = sg/16, s = sg%16, u = e%100.
    float creg[13];
    int   gidx[13];   // pair*G_SZ + s*HP + u   (offset within G region)
    int   hidx[13];   // pair*HA_SZ + hA_addr(s,u)  (offset within HA region)
    int   lenr[13];   // sequence length (0 for padding slots -> never commit)
#pragma unroll
    for (int r = 0; r < 13; ++r) {
        int e = tid + r * 256;
        creg[r] = 0.0f;
        if (e < SEQ_PER_WG * HIDDEN) {
            int sg = e / HIDDEN;
            int u  = e - sg * HIDDEN;
            int p  = sg >> 4, s = sg & 15;
            gidx[r] = p * G_SZ + s * HP + u;
            hidx[r] = p * HA_SZ + hA_addr(s, u);
            lenr[r] = slen[sg];
        } else {
            gidx[r] = 0; hidx[r] = 0; lenr[r] = 0;
        }
    }
    __syncthreads();

    // ---------------- time recurrence ----------------
    for (int t = 0; t < maxlen; ++t) {
        // ---- gates GEMM: wave (pair,gate) -> tiles gate*7 .. gate*7+6 ----
        v8f acc[TPW];
        {   // K-iter 0 peeled: C = 0 lowers to the inline-zero operand.
            v2f ah = *(const v2f*)(hAp + lane * 2);
#pragma unroll
            for (int tt = 0; tt < TPW; ++tt) {
                int tg = wave * TPW + tt;
                v2f bw = *(const v2f*)(wB + (tg * 32 + lane) * 2);
                acc[tt] = __builtin_amdgcn_wmma_f32_16x16x4_f32(
                    false, ah, false, bw, (short)0,
                    (v8f){0.f, 0.f, 0.f, 0.f, 0.f, 0.f, 0.f, 0.f}, false, false);
            }
        }
        for (int k4 = 1; k4 < KITERS; ++k4) {
            v2f ah = *(const v2f*)(hAp + (k4 * 32 + lane) * 2);
#pragma unroll
            for (int tt = 0; tt < TPW; ++tt) {
                int tg = wave * TPW + tt;
                v2f bw = *(const v2f*)(wB + ((k4 * NTILES + tg) * 32 + lane) * 2);
                acc[tt] = __builtin_amdgcn_wmma_f32_16x16x4_f32(
                    false, ah, false, bw, (short)0, acc[tt], false, false);
            }
        }

        // ---- activate + publish gates (wave-uniform branch) ----
        // D layout: lane holds col = 16*tile + lane%16; VGPR v holds row
        // m = v + 8*(lane/16).
        {
            const int u0    = lane & 15;
            const int sbase = (lane >> 4) * 8;
            if (wave == 2) {
#pragma unroll
                for (int tt = 0; tt < TPW; ++tt) {
                    int u = tt * 16 + u0;
#pragma unroll
                    for (int v = 0; v < 8; ++v)
                        Gp[(2 * 16 + sbase + v) * HP + u] =
                            tanh_fast(acc[tt][v]);
                }
            } else {
#pragma unroll
                for (int tt = 0; tt < TPW; ++tt) {
                    int u = tt * 16 + u0;
#pragma unroll
                    for (int v = 0; v < 8; ++v)
                        Gp[(wave * 16 + sbase + v) * HP + u] =
                            sigmoid_fast(acc[tt][v]);
                }
            }
        }
        __syncthreads();

        // ---- elementwise cell update (precomputed indices, no div/mod) ----
        {
            float* Gb  = smem + G_OFF;
            float* hAb = smem + HA_OFF;
#pragma unroll
            for (int r = 0; r < 13; ++r) {
                float iv = Gb[0 * 16 * HP + gidx[r]];
                float fv = Gb[1 * 16 * HP + gidx[r]];
                float gv = Gb[2 * 16 * HP + gidx[r]];
                float ov = Gb[3 * 16 * HP + gidx[r]];
                float cn = fv * creg[r] + iv * gv;
                float hn = ov * tanh_fast(cn);
                if (t < lenr[r]) {               // freeze h, c past length
                    creg[r] = cn;
                    hAb[hidx[r]] = hn;
                }
            }
        }

        // ---- stage x_{t+1} into the augmented state slots ----
        if (tid < SEQ_PER_WG && (t + 1) < maxlen) {
            int p = tid >> 4, s = tid & 15;
            float* hAq = smem + HA_OFF + p * HA_SZ;
            v4f xv = *(const v4f*)(x + ((size_t)(seq_base + tid) * T_MAX + (t + 1))
                                       * INPUT);
            hAq[(26 * 32 + s) * 2 + 0]        = xv.x;
            hAq[(26 * 32 + s) * 2 + 1]        = xv.y;
            hAq[(26 * 32 + (s + 16)) * 2 + 0] = xv.z;
            hAq[(26 * 32 + (s + 16)) * 2 + 1] = xv.w;
        }
        __syncthreads();
    }

    // ---------------- epilogue: final h (frozen at last valid step) --------
    for (int e = tid; e < SEQ_PER_WG * HIDDEN; e += 256) {
        int sg = e / HIDDEN;
        int u  = e - sg * HIDDEN;
        int p  = sg >> 4, s = sg & 15;
        out[(size_t)(seq_base + sg) * HIDDEN + u] =
            smem[HA_OFF + p * HA_SZ + hA_addr(s, u)];
    }
}

extern "C" void kernel_launch(void* const* d_in, const int* in_sizes, int n_in,
                              void* d_out, int out_size, void* d_ws, size_t ws_size,
                              hipStream_t stream) {
    (void)in_sizes; (void)n_in; (void)out_size; (void)d_ws; (void)ws_size;
    const float* x    = (const float*)d_in[0];
    const float* w_ih = (const float*)d_in[1];
    const float* w_hh = (const float*)d_in[2];
    const float* b_ih = (const float*)d_in[3];
    const float* b_hh = (const float*)d_in[4];
    const int*   len  = (const int*)d_in[5];
    float* out = (float*)d_out;

    const size_t shmem = (size_t)SMEM_FLOATS * sizeof(float);  // 264832 B
    hipFuncSetAttribute((const void*)LongTermLSTM_40389872452133_kernel,
                        hipFuncAttributeMaxDynamicSharedMemorySize, (int)shmem);

    dim3 grid(4096 / SEQ_PER_WG);  // 128 workgroups, 32 sequences each
    dim3 block(256);               // 8 waves: {pair 0,1} x {gates i,f,g,o}
    LongTermLSTM_40389872452133_kernel<<<grid, block, shmem, stream>>>(
        x, w_ih, w_hh, b_ih, b_hh, len, out);
}